// SSFM_48266842472970
// MI455X (gfx1250) — compile-verified
//
#include <hip/hip_runtime.h>
#include <math.h>
#include <stdint.h>

// ---------------------------------------------------------------------------
// SSFM (nonlinear Schroedinger split-step) for MI455X / gfx1250.
// FFT-65536 as 256x256 four-step; each 256-pt FFT done as two complex
// 16x16x16 matmuls on the FP32 WMMA pipe (v_wmma_f32_16x16x4_f32),
// one FFT per wave32, fragments staged through LDS.
// Round 2: DFT/twiddle factors from per-block LDS tables (kills the sincos
// storm seen in round 1), async global->LDS fill in the inverse-column pass.
// ---------------------------------------------------------------------------

typedef __attribute__((ext_vector_type(2))) float v2f;
typedef __attribute__((ext_vector_type(8))) float v8f;

#define NTOT   65536
#define NBATCH 32
#define PI2f   6.28318530717958647692f

#define USE_ASYNC_LDS 1

// ---- physics constants (match reference) ----
#define ALPHA_f   0.046f
#define BETA2_f  -21.67f
#define FC_f      193.55f
#define GAMMA_f   1.27f
#define DZ_f      2.5f       // ELL / N_SSFM = 80/32
#define DZH_f     1.25f
#define HWL_i     100

__device__ __forceinline__ float noise_std() {
  // sqrt( (h/T^2 * 2*pi*Fc*alpha*ell*nsp) / dt / N_ssfm / 2 )
  const float rho = 6.62607015e-10f * PI2f * FC_f * ALPHA_f * 80.0f;
  return __builtin_sqrtf(rho / 32.0f / 2.0f);
}

// ---- FP32 WMMA 16x16x4:  D = A(16x4) * B(4x16) + C ----
__device__ __forceinline__ v8f wmma4(v2f a, v2f b, v8f c) {
  return __builtin_amdgcn_wmma_f32_16x16x4_f32(
      /*neg_a=*/false, a, /*neg_b=*/false, b,
      /*c_mod=*/(short)0, c, /*reuse_a=*/false, /*reuse_b=*/false);
}

// ---------------------------------------------------------------------------
// Per-block twiddle tables (computed once, consumed via ds_load broadcasts).
// w16[t]  = W16^t  = exp(-2pi i t/16)   (16 distinct DFT-matrix values)
// w256[t] = W256^t = exp(-2pi i t/256)  (inter-stage twiddles)
// Inverse transform = conjugate = flip sign of the imaginary part.
// ---------------------------------------------------------------------------
__device__ __forceinline__ void build_tables(float2* w16, float2* w256) {
  if (threadIdx.x < 16) {
    float s, c;
    __sincosf(-(PI2f / 16.0f) * (float)threadIdx.x, &s, &c);
    w16[threadIdx.x] = make_float2(c, s);
  }
  for (int t = threadIdx.x; t < 256; t += 128) {
    float s, c;
    __sincosf(-(PI2f / 256.0f) * (float)t, &s, &c);
    w256[t] = make_float2(c, s);
  }
  __syncthreads();
}

// ---------------------------------------------------------------------------
// 256-point complex FFT executed by ONE wave32 entirely on WMMA.
// data: 256 complex in natural order (in & out). tmp: 256 complex scratch.
// Decomposition: x[n1+16*n2] -> X[k2+16*k1];   O = F16 * V, twiddle, * F16.
// A-frag layout (16x4 fp32): lanes0-15 row=lane K={4c,4c+1}; lanes16-31 K={4c+2,4c+3}.
// B-frag layout (4x16):      v0: K=4c(+2 hi-half), v1: K=4c+1(+3); col=lane&15.
// C/D layout: vgpr r -> row r (lanes0-15) / r+8 (lanes16-31), col=lane&15.
// ---------------------------------------------------------------------------
__device__ __forceinline__ void wave_fft256(float2* data, float2* tmp,
                                            const float2* w16,
                                            const float2* w256,
                                            float sgn, int lane) {
  const int half = (lane < 16) ? 0 : 1;
  const int l15  = lane & 15;

  // ---- Stage 1: S = F16 x V  (V[n2][n1] = data[n1 + 16*n2]) ----
  v8f dr = {}; v8f di = {};
  for (int c = 0; c < 4; ++c) {
    const int ak = 4 * c + 2 * half;
    float2 f0 = w16[(l15 * ak) & 15];          // F[l15][ak]
    float2 f1 = w16[(l15 * (ak + 1)) & 15];
    v2f fr  = {f0.x, f1.x};
    v2f fi  = {sgn * f0.y, sgn * f1.y};
    v2f fin = {-fi.x, -fi.y};
    float2 b0 = data[l15 + 16 * ak];
    float2 b1 = data[l15 + 16 * (ak + 1)];
    v2f vr = {b0.x, b1.x};
    v2f vi = {b0.y, b1.y};
    dr = wmma4(fr,  vr, dr);   // Re += Fr*Vr
    dr = wmma4(fin, vi, dr);   // Re -= Fi*Vi
    di = wmma4(fr,  vi, di);   // Im += Fr*Vi
    di = wmma4(fi,  vr, di);   // Im += Fi*Vr
  }

  // ---- Twiddle W256^{k2*n1} and park T row-major in tmp ----
  for (int r = 0; r < 8; ++r) {
    const int M  = r + 8 * half;   // k2
    const int Nn = l15;            // n1
    float2 t = w256[(M * Nn) & 255];
    float twc = t.x, tws = sgn * t.y;
    float xr = dr[r], xi = di[r];
    tmp[Nn + 16 * M] = make_float2(xr * twc - xi * tws, xr * tws + xi * twc);
  }
  __syncthreads();

  // ---- Stage 2: O = T x F16 ----
  v8f er = {}; v8f ei = {};
  for (int c = 0; c < 4; ++c) {
    const int ak = 4 * c + 2 * half;
    float2 a0 = tmp[ak + 16 * l15];            // T[row=l15][K=ak]
    float2 a1 = tmp[(ak + 1) + 16 * l15];
    v2f tr  = {a0.x, a1.x};
    v2f ti  = {a0.y, a1.y};
    float2 f0 = w16[(ak * l15) & 15];          // F[ak][l15] (symmetric)
    float2 f1 = w16[((ak + 1) * l15) & 15];
    v2f fr  = {f0.x, f1.x};
    v2f fi  = {sgn * f0.y, sgn * f1.y};
    v2f fin = {-fi.x, -fi.y};
    er = wmma4(tr, fr,  er);
    er = wmma4(ti, fin, er);
    ei = wmma4(tr, fi,  ei);
    ei = wmma4(ti, fr,  ei);
  }
  __syncthreads();

  // ---- Unscramble: O[k2][k1] = X[k2 + 16*k1] -> natural order ----
  for (int r = 0; r < 8; ++r) {
    const int M  = r + 8 * half;   // k2
    const int Nn = l15;            // k1
    data[M + 16 * Nn] = make_float2(er[r], ei[r]);
  }
  __syncthreads();
}

// ---- deterministic counter-based Gaussian (hash -> Box-Muller) ----
__device__ __forceinline__ uint32_t mix32(uint32_t x) {
  x ^= x >> 16; x *= 0x7feb352du;
  x ^= x >> 15; x *= 0x846ca68bu;
  x ^= x >> 16; return x;
}
__device__ __forceinline__ float2 gauss2(uint32_t step, uint32_t idx) {
  uint32_t h1 = mix32(idx * 0x9E3779B9u + step * 0x85EBCA6Bu + 0x68bc21ebu);
  uint32_t h2 = mix32(h1 ^ 0x2545F491u);
  float u1 = ((float)(h1 >> 8) + 1.0f) * (1.0f / 16777216.0f);  // (0,1]
  float u2 = (float)(h2 >> 8) * (1.0f / 16777216.0f);
  float r  = __builtin_sqrtf(-2.0f * __logf(u1));
  float s, c; __sincosf(PI2f * u2, &s, &c);
  return make_float2(r * c, r * s);
}

__device__ __forceinline__ float hamm(int n) {
  if (n < HWL_i)
    return 0.54f - 0.46f * __cosf((PI2f / 200.0f) * (float)n);
  if (n >= NTOT - HWL_i)
    return 0.54f - 0.46f * __cosf((PI2f / 200.0f) * (float)(n - NTOT + 200));
  return 1.0f;
}

// ===========================================================================
// Kernel 0: pack planar (re,im) -> interleaved complex state
// ===========================================================================
__global__ void k_pack(const float* __restrict__ re, const float* __restrict__ im,
                       float2* __restrict__ q) {
  int i = blockIdx.x * blockDim.x + threadIdx.x;
  q[i] = make_float2(re[i], im[i]);
}

// ===========================================================================
// Kernel 1 (forward columns): fused window/Kerr/noise, FFT-256 over n2,
// outer twiddle W_N^{n1*k2}, write Y[b][n1*256 + k2] (contiguous).
// One FFT per wave; 4 waves/block; 2048 blocks = 32 batches x 256 columns.
// ===========================================================================
__global__ void __launch_bounds__(128) k_col(const float2* __restrict__ q,
                                             float2* __restrict__ y,
                                             int applyWin, int applyNL,
                                             int noiseStep) {
  const int wave = threadIdx.x >> 5, lane = threadIdx.x & 31;
  const int inst = blockIdx.x * 4 + wave;
  const int b = inst >> 8, n1 = inst & 255;
  __shared__ float2 sd[4][256], st[4][256];
  __shared__ float2 w16t[16], w256t[256];
  build_tables(w16t, w256t);
  float2* D = sd[wave];
  float2* T = st[wave];

  const float2* src = q + (size_t)b * NTOT;
  const float nstd = noise_std();
  for (int j = lane; j < 256; j += 32) {          // j = n2
    const int n = n1 + (j << 8);                  // time index
    float2 v = src[n];
    if (applyWin) { float w = hamm(n); v.x *= w; v.y *= w; }
    if (applyNL) {
      float p = -GAMMA_f * DZ_f * (v.x * v.x + v.y * v.y);
      float s, c; __sincosf(p, &s, &c);
      v = make_float2(v.x * c - v.y * s, v.x * s + v.y * c);
    }
    if (noiseStep >= 0) {
      float2 nz = gauss2((uint32_t)noiseStep, (uint32_t)(b * NTOT + n));
      v.x += nz.x * nstd; v.y += nz.y * nstd;
    }
    D[j] = v;
  }
  __syncthreads();

  wave_fft256(D, T, w16t, w256t, 1.0f, lane);     // n2 -> k2

  float2* dst = y + (size_t)b * NTOT + (size_t)n1 * 256;
  for (int k2 = lane; k2 < 256; k2 += 32) {
    float s, c;
    __sincosf((-PI2f / 65536.0f) * (float)(n1 * k2), &s, &c);
    float2 v = D[k2];
    dst[k2] = make_float2(v.x * c - v.y * s, v.x * s + v.y * c);
  }
}

// ===========================================================================
// Kernel 2 (middle, in place per column k2): FFT-256 over n1 -> k1,
// dispersion multiply at k = k2 + 256*k1 (fftshift == k ^ 0x8000),
// inverse FFT-256 over k1 -> n1, conj outer twiddle, write back.
// ===========================================================================
__global__ void __launch_bounds__(128) k_mid(float2* __restrict__ y, float dzLin) {
  const int wave = threadIdx.x >> 5, lane = threadIdx.x & 31;
  const int inst = blockIdx.x * 4 + wave;
  const int b = inst >> 8, k2 = inst & 255;
  __shared__ float2 sd[4][256], st[4][256];
  __shared__ float2 w16t[16], w256t[256];
  build_tables(w16t, w256t);
  float2* D = sd[wave];
  float2* T = st[wave];

  float2* base = y + (size_t)b * NTOT;
  for (int j = lane; j < 256; j += 32)            // j = n1 (stride-256 gather)
    D[j] = base[(size_t)j * 256 + k2];
  __syncthreads();

  wave_fft256(D, T, w16t, w256t, 1.0f, lane);     // n1 -> k1

  const float thc = (-BETA2_f * 0.5f) * dzLin;    // exp(i * thc * (2*pi*f)^2)
  for (int k1 = lane; k1 < 256; k1 += 32) {
    const int k  = k2 + (k1 << 8);
    const int jf = k ^ 32768;                     // fftshift index
    float f  = ((float)jf - 32768.0f) * (1.0f / 65536.0f);
    float w  = PI2f * f;
    float th = thc * w * w;
    float s, c; __sincosf(th, &s, &c);
    float2 v = D[k1];
    D[k1] = make_float2(v.x * c - v.y * s, v.x * s + v.y * c);
  }
  __syncthreads();

  wave_fft256(D, T, w16t, w256t, -1.0f, lane);    // k1 -> n1 (inverse, unscaled)

  for (int n1 = lane; n1 < 256; n1 += 32) {
    float s, c;
    __sincosf((PI2f / 65536.0f) * (float)(n1 * k2), &s, &c);   // conj twiddle
    float2 v = D[n1];
    base[(size_t)n1 * 256 + k2] =
        make_float2(v.x * c - v.y * s, v.x * s + v.y * c);
  }
}

// ===========================================================================
// Kernel 3 (inverse columns): inverse FFT-256 over k2 -> n2, scale 1/N,
// scatter to out[b][n1 + 256*n2]. Row fill uses CDNA5 async global->LDS
// (ASYNCcnt-tracked, no VGPR transit), then s_wait_asynccnt + barrier.
// ===========================================================================
__global__ void __launch_bounds__(128) k_coli(const float2* __restrict__ y,
                                              float2* __restrict__ out) {
  const int wave = threadIdx.x >> 5, lane = threadIdx.x & 31;
  const int inst = blockIdx.x * 4 + wave;
  const int b = inst >> 8, n1 = inst & 255;
  __shared__ float2 sd[4][256], st[4][256];
  __shared__ float2 w16t[16], w256t[256];
  build_tables(w16t, w256t);
  float2* D = sd[wave];
  float2* T = st[wave];

  const float2* row = y + (size_t)b * NTOT + (size_t)n1 * 256;
#if USE_ASYNC_LDS
  for (int j = lane; j < 256; j += 32) {          // j = k2
    // LDS byte offset = low 32 bits of the generic LDS pointer on AMDGPU.
    uint32_t ldsOff = (uint32_t)(uintptr_t)(&D[j]);
    unsigned long long gaddr = (unsigned long long)(uintptr_t)(row + j);
    asm volatile("global_load_async_to_lds_b64 %0, %1, off"
                 :: "v"(ldsOff), "v"(gaddr) : "memory");
  }
  asm volatile("s_wait_asynccnt 0x0" ::: "memory");
#else
  for (int j = lane; j < 256; j += 32) D[j] = row[j];
#endif
  __syncthreads();

  wave_fft256(D, T, w16t, w256t, -1.0f, lane);    // k2 -> n2 (inverse)

  const float sc = 1.0f / 65536.0f;
  float2* dst = out + (size_t)b * NTOT;
  for (int n2 = lane; n2 < 256; n2 += 32) {
    float2 v = D[n2];
    dst[n1 + (n2 << 8)] = make_float2(v.x * sc, v.y * sc);
  }
}

// ===========================================================================
extern "C" void kernel_launch(void* const* d_in, const int* in_sizes, int n_in,
                              void* d_out, int out_size, void* d_ws, size_t ws_size,
                              hipStream_t stream) {
  (void)in_sizes; (void)n_in; (void)out_size; (void)ws_size;
  const float* xr = (const float*)d_in[0];
  const float* xi = (const float*)d_in[1];
  float2* qA = (float2*)d_ws;                     // 16 MB state
  float2* Y  = qA + (size_t)NBATCH * NTOT;        // 16 MB scratch
  float2* out = (float2*)d_out;                   // interleaved complex64

  const int NB = (NBATCH * 256) / 4;              // 2048 blocks, 4 waves each

  k_pack<<<(NBATCH * NTOT) / 256, 256, 0, stream>>>(xr, xi, qA);

  auto linear = [&](float2* srcDst, float2* finalDst, float dzLin,
                    int win, int nl, int noise) {
    k_col <<<NB, 128, 0, stream>>>(srcDst, Y, win, nl, noise);
    k_mid <<<NB, 128, 0, stream>>>(Y, dzLin);
    k_coli<<<NB, 128, 0, stream>>>(Y, finalDst);
  };

  // initial half linear step, no pointwise pre-ops
  linear(qA, qA, DZH_f, 0, 0, -1);
  // 31 full steps: window -> Kerr -> noise -> full linear
  for (int s = 0; s < 31; ++s) linear(qA, qA, DZ_f, 1, 1, s);
  // final: Kerr -> noise (no window) -> half linear, straight into d_out
  linear(qA, out, DZH_f, 0, 1, 31);
}